// SSTModel_72541997630132
// MI455X (gfx1250) — compile-verified
//
#include <hip/hip_runtime.h>

// Problem constants (from reference: coeffs [16, 256, 8192] f32)
constexpr int kB = 16;
constexpr int kF = 256;
constexpr int kT = 8192;

// Tile shape: FT output rows x TT output cols per block
constexpr int kFT = 16;
constexpr int kTT = 256;
constexpr int kLRows = kFT + 2;   // halo rows f0-1 .. f0+FT
constexpr int kLW    = kTT + 4;   // 260 cols: +1 needed for t+1, rounded to float4
constexpr int kChunks = kLW / 4;              // 65 float4 chunks per row
constexpr int kTotalChunks = kLRows * kChunks; // 1170

#if defined(__gfx1250__) && __has_builtin(__builtin_amdgcn_global_load_async_to_lds_b128)
#define SST_ASYNC_LDS 1
#else
#define SST_ASYNC_LDS 0
#endif

typedef int v4i __attribute__((ext_vector_type(4)));
typedef __attribute__((address_space(1))) v4i* gv4i_ptr;  // global (AS1)
typedef __attribute__((address_space(3))) v4i* lv4i_ptr;  // LDS (AS3)

__global__ __launch_bounds__(256) void sst_scatter_stencil_kernel(
    const float* __restrict__ x, float* __restrict__ out) {
  __shared__ float tile[kLRows * kLW];

  const int tid = threadIdx.x;
  const int t0  = blockIdx.x * kTT;
  const int f0  = blockIdx.y * kFT;
  const int b   = blockIdx.z;
  const long long base = (long long)b * kF * kT;

  // ---- Stage (kLRows x kLW) halo tile into LDS ------------------------------
  // Addresses are clamped to stay in-bounds; clamped (garbage) values are only
  // ever masked out by the predicates in the compute phase.
  for (int i = tid; i < kTotalChunks; i += 256) {
    const int r  = i / kChunks;
    const int cc = i - r * kChunks;
    int f = f0 - 1 + r;
    f = f < 0 ? 0 : (f > kF - 1 ? kF - 1 : f);
    int t = t0 + cc * 4;
    t = t > kT - 4 ? kT - 4 : t;
    const float* src = x + base + (long long)f * kT + t;
    float* dst = &tile[r * kLW + cc * 4];
#if SST_ASYNC_LDS
    __builtin_amdgcn_global_load_async_to_lds_b128(
        (gv4i_ptr)const_cast<float*>(src), (lv4i_ptr)dst, 0, 0);
#else
    *(float4*)dst = *(const float4*)src;
#endif
  }
#if SST_ASYNC_LDS
#if __has_builtin(__builtin_amdgcn_s_wait_asynccnt)
  __builtin_amdgcn_s_wait_asynccnt(0);
#else
  asm volatile("s_wait_asynccnt 0" ::: "memory");
#endif
#endif
  __syncthreads();

  // ---- Compute: out[b,g,t] = sum of up-to-3 predicated neighbor values ------
  const int col  = (tid & 63) * 4;   // 0..252
  const int row0 = (tid >> 6) * 4;   // 0,4,8,12

#pragma unroll
  for (int rr = 0; rr < 4; ++rr) {
    const int row = row0 + rr;
    const int g   = f0 + row;
    const int lr  = row + 1;  // LDS row of f == g
    const float* Lc = &tile[lr * kLW + col];
    const float* Ll = &tile[(lr - 1) * kLW + col];
    const float* Lr = &tile[(lr + 1) * kLW + col];

    float4 o;
    float* op = reinterpret_cast<float*>(&o);
#pragma unroll
    for (int e = 0; e < 4; ++e) {
      const int t = t0 + col + e;
      const bool notLast = (t < kT - 1);

      const float xc = Lc[e];
      const float xl = Ll[e];
      const float xr = Lr[e];

      // phase/pi in {0,1} via sign bit (matches arctan2(0,x): pi iff signbit)
      const int pc  = (int)(__float_as_uint(xc)      >> 31);
      const int pcn = (int)(__float_as_uint(Lc[e+1]) >> 31);
      const int pl  = (int)(__float_as_uint(xl)      >> 31);
      const int pln = (int)(__float_as_uint(Ll[e+1]) >> 31);
      const int pr  = (int)(__float_as_uint(xr)      >> 31);
      const int prn = (int)(__float_as_uint(Lr[e+1]) >> 31);

      const int d  = notLast ? (pcn - pc) : 0;
      const int dl = notLast ? (pln - pl) : 0;
      const int dr = notLast ? (prn - pr) : 0;

      // center: stays at k==g if delta==0, or clamped at the edges
      float acc = (d == 0 || (g == 0 && d < 0) || (g == kF - 1 && d > 0)) ? xc : 0.0f;
      // left neighbor f = g-1 lands on g when its delta == +1
      acc += (g > 0 && dl == 1) ? xl : 0.0f;
      // right neighbor f = g+1 lands on g when its delta == -1
      acc += (g < kF - 1 && dr == -1) ? xr : 0.0f;
      op[e] = acc;
    }
    float* dst = out + base + (long long)g * kT + t0 + col;
    *(float4*)dst = o;
  }
}

extern "C" void kernel_launch(void* const* d_in, const int* in_sizes, int n_in,
                              void* d_out, int out_size, void* d_ws, size_t ws_size,
                              hipStream_t stream) {
  (void)in_sizes; (void)n_in; (void)out_size; (void)d_ws; (void)ws_size;
  const float* x = (const float*)d_in[0];
  float* out = (float*)d_out;
  dim3 grid(kT / kTT, kF / kFT, kB);  // 32 x 16 x 16 = 8192 blocks
  sst_scatter_stencil_kernel<<<grid, 256, 0, stream>>>(x, out);
}